// MambaLayer_11476152615153
// MI455X (gfx1250) — compile-verified
//
#include <hip/hip_runtime.h>
#include <hip/hip_bf16.h>
#include <math.h>

#define D_MODEL 768
#define D_STATE 64
#define D_CONV 4
#define D_INNER 1536
#define HEADDIM 192
#define NHEADS 8
#define CONV_DIM 1664            // D_INNER + 2*D_STATE
#define D_IN_PROJ 3208           // 2*D_INNER + 2*D_STATE + NHEADS
#define BATCH 2
#define SEQLEN 2048
#define ROWS (BATCH * SEQLEN)    // 4096
#define NSPLIT 4                 // scan: n-dimension split (64/NSPLIT states per thread)

typedef float v2f __attribute__((ext_vector_type(2)));
typedef float v8f __attribute__((ext_vector_type(8)));

__device__ __forceinline__ float silu_f(float x) { return x / (1.f + __expf(-x)); }
__device__ __forceinline__ float softplus_f(float x) { return x > 20.f ? x : log1pf(__expf(x)); }

// ---- block reduction (blockDim.x == 256, wave32) ----
__device__ __forceinline__ float block_reduce_sum(float v, float* sm) {
    int tid = threadIdx.x;
    #pragma unroll
    for (int off = 16; off; off >>= 1) v += __shfl_xor(v, off, 32);
    if ((tid & 31) == 0) sm[tid >> 5] = v;
    __syncthreads();
    float r = (tid < 8) ? sm[tid] : 0.f;
    if (tid < 32) {
        #pragma unroll
        for (int off = 4; off; off >>= 1) r += __shfl_xor(r, off, 32);
        if (tid == 0) sm[0] = r;
    }
    __syncthreads();
    float out = sm[0];
    __syncthreads();
    return out;
}

// ---- 1) LayerNorm over D_MODEL ----
__global__ void __launch_bounds__(256)
ln_kernel(const float* __restrict__ x, const float* __restrict__ w,
          const float* __restrict__ b, float* __restrict__ out) {
    __shared__ float sm[8];
    int row = blockIdx.x;
    const float* xr = x + (size_t)row * D_MODEL;
    float v[3];
    float s = 0.f;
    #pragma unroll
    for (int i = 0; i < 3; ++i) { v[i] = xr[threadIdx.x + i * 256]; s += v[i]; }
    float mean = block_reduce_sum(s, sm) * (1.f / D_MODEL);
    float s2 = 0.f;
    #pragma unroll
    for (int i = 0; i < 3; ++i) { float d = v[i] - mean; s2 += d * d; }
    float var = block_reduce_sum(s2, sm) * (1.f / D_MODEL);
    float inv = rsqrtf(var + 1e-12f);
    #pragma unroll
    for (int i = 0; i < 3; ++i) {
        int c = threadIdx.x + i * 256;
        out[(size_t)row * D_MODEL + c] = (v[i] - mean) * inv * w[c] + b[c];
    }
}

// helper: one K4 WMMA group over the 4 N-tiles
#define WMMA_GROUP(AV, BV, ACC)                                             \
    _Pragma("unroll")                                                        \
    for (int t = 0; t < 4; ++t)                                              \
        ACC[t] = __builtin_amdgcn_wmma_f32_16x16x4_f32(                      \
            false, AV, false, BV[t], (short)0, ACC[t], false, false);

#define LOAD_CHUNK(A0, A1, B0, B1, KOFF)                                     \
    do {                                                                     \
        A0 = *(const v2f*)(Arow + (KOFF));                                   \
        A1 = *(const v2f*)(Arow + (KOFF) + 4);                               \
        _Pragma("unroll")                                                    \
        for (int t = 0; t < 4; ++t) {                                        \
            B0[t] = *(const v2f*)(Brow[t] + (KOFF));                         \
            B1[t] = *(const v2f*)(Brow[t] + (KOFF) + 4);                     \
        }                                                                    \
    } while (0)

// ---- 2/6) fp32 WMMA GEMM:  C[M x N] = A[M x K] * W[N x K]^T (+ resid) ----
// One wave -> 16(M) x 64(N) tile. Ping-pong pipeline over K chunks of 8
// (two 16x16x4 WMMA groups per chunk): compute set X, refill X; compute set
// Y, refill Y. No pipeline-register copies, loads stay a half-iteration in
// flight ahead of their consuming WMMAs. K must be a multiple of 16.
// grid.x = ceil(N/64), grid.y = M/128, block = 256 threads (8 waves).
__global__ void __launch_bounds__(256)
wmma_gemm_f32(const float* __restrict__ A, const float* __restrict__ W,
              float* __restrict__ C, const float* __restrict__ resid,
              int K, int N, int ldc) {
    int wave  = threadIdx.x >> 5;
    int lane  = threadIdx.x & 31;
    int lhalf = lane >> 4;       // selects K pair (A/B frag) and M+8 (D frag)
    int lidx  = lane & 15;       // M index for A frag, N index for B frag
    int m0 = blockIdx.y * 128 + wave * 16;
    int n0 = blockIdx.x * 64;

    const float* Arow = A + (size_t)(m0 + lidx) * K + 2 * lhalf;
    const float* Brow[4];
    int ncol[4];
    #pragma unroll
    for (int t = 0; t < 4; ++t) {
        int j = n0 + t * 16 + lidx;
        ncol[t] = j;
        int jc = (j < N) ? j : (N - 1);      // clamp tail loads; stores are guarded
        Brow[t] = W + (size_t)jc * K + 2 * lhalf;
    }

    v8f acc[4];
    #pragma unroll
    for (int t = 0; t < 4; ++t)
        #pragma unroll
        for (int r = 0; r < 8; ++r) acc[t][r] = 0.f;

    // two pipeline stages: X (chunk k) and Y (chunk k+8)
    v2f aX0, aX1, bX0[4], bX1[4];
    v2f aY0, aY1, bY0[4], bY1[4];
    LOAD_CHUNK(aX0, aX1, bX0, bX1, 0);
    LOAD_CHUNK(aY0, aY1, bY0, bY1, 8);

    for (int k = 16; k + 16 <= K; k += 16) {
        WMMA_GROUP(aX0, bX0, acc);
        WMMA_GROUP(aX1, bX1, acc);
        LOAD_CHUNK(aX0, aX1, bX0, bX1, k);          // refill X
        WMMA_GROUP(aY0, bY0, acc);
        WMMA_GROUP(aY1, bY1, acc);
        LOAD_CHUNK(aY0, aY1, bY0, bY1, k + 8);      // refill Y
    }
    // epilogue: last two chunks already resident in X and Y
    WMMA_GROUP(aX0, bX0, acc);
    WMMA_GROUP(aX1, bX1, acc);
    WMMA_GROUP(aY0, bY0, acc);
    WMMA_GROUP(aY1, bY1, acc);

    #pragma unroll
    for (int t = 0; t < 4; ++t) {
        if (ncol[t] < N) {
            #pragma unroll
            for (int r = 0; r < 8; ++r) {
                size_t o = (size_t)(m0 + r + 8 * lhalf) * ldc + ncol[t];
                float v = acc[t][r];
                if (resid) v += resid[o];
                C[o] = v;
            }
        }
    }
}

// ---- 3) depthwise causal conv(4) + SiLU on xBC channels, softplus(dt) ----
__global__ void __launch_bounds__(256)
conv_dt_kernel(const float* __restrict__ zx, const float* __restrict__ cw,
               const float* __restrict__ cb, const float* __restrict__ dt_bias,
               float* __restrict__ xbc, float* __restrict__ dtw) {
    long idx = (long)blockIdx.x * 256 + threadIdx.x;
    const int CPR = CONV_DIM + NHEADS;           // 1672 work items per row
    long total = (long)ROWS * CPR;
    if (idx >= total) return;
    int row = (int)(idx / CPR);
    int c   = (int)(idx % CPR);
    int l   = row & (SEQLEN - 1);
    if (c < CONV_DIM) {
        float acc = cb[c];
        #pragma unroll
        for (int t = 0; t < D_CONV; ++t) {
            int ls = l - (D_CONV - 1) + t;
            if (ls >= 0)
                acc += zx[(size_t)(row - (D_CONV - 1) + t) * D_IN_PROJ + D_INNER + c] * cw[c * D_CONV + t];
        }
        xbc[(size_t)row * CONV_DIM + c] = silu_f(acc);
    } else {
        int h = c - CONV_DIM;
        dtw[(size_t)row * NHEADS + h] =
            softplus_f(zx[(size_t)row * D_IN_PROJ + D_INNER + CONV_DIM + h] + dt_bias[h]);
    }
}

// ---- 4) selective scan: one block per (batch, head) ----
// 768 threads = 192 p-values x 4 n-quarters (16 states each, in registers).
// 24 wave32 per block x 16 blocks = 384 waves to hide ds/exp latency on the
// serial critical path. Partial y combined through LDS, 2 barriers per step.
__global__ void __launch_bounds__(NSPLIT * HEADDIM)
scan_kernel(const float* __restrict__ xbc, const float* __restrict__ dtw,
            const float* __restrict__ A_log, const float* __restrict__ Dp,
            float* __restrict__ yraw) {
    const int NPT = D_STATE / NSPLIT;   // 16 states per thread
    int b = blockIdx.x >> 3;
    int h = blockIdx.x & 7;
    int tid = threadIdx.x;
    int q   = tid / HEADDIM;            // n-quarter 0..3
    int p   = tid % HEADDIM;
    int n0  = q * NPT;
    float A  = -__expf(A_log[h]);
    float Dv = Dp[h];

    __shared__ float sB[D_STATE], sC[D_STATE], sYp[NSPLIT - 1][HEADDIM];
    float hs[NPT];
    #pragma unroll
    for (int i = 0; i < NPT; ++i) hs[i] = 0.f;

    for (int l = 0; l < SEQLEN; ++l) {
        size_t row = (size_t)b * SEQLEN + l;
        const float* xr = xbc + row * CONV_DIM;
        if (tid < D_STATE)            sB[tid] = xr[D_INNER + tid];
        else if (tid < 2 * D_STATE)   sC[tid - D_STATE] = xr[D_INNER + D_STATE + (tid - D_STATE)];
        float dt = dtw[row * NHEADS + h];
        float x  = xr[h * HEADDIM + p];
        __syncthreads();

        float dA  = __expf(dt * A);
        float xdt = x * dt;
        float yp  = 0.f;
        #pragma unroll
        for (int i = 0; i < NPT; ++i) {
            hs[i] = hs[i] * dA + xdt * sB[n0 + i];
            yp   += hs[i] * sC[n0 + i];
        }
        if (q) sYp[q - 1][p] = yp;
        __syncthreads();
        if (q == 0) {
            float y = yp + sYp[0][p] + sYp[1][p] + sYp[2][p] + Dv * x;
            yraw[row * D_INNER + h * HEADDIM + p] = y;
        }
        // next-iter sB/sC staging doesn't alias sYp; 2 barriers/step suffice
    }
}

// ---- 5) SiLU(z) gate + RMSNorm ----
__global__ void __launch_bounds__(256)
gate_rms_kernel(const float* __restrict__ yraw, const float* __restrict__ zx,
                const float* __restrict__ rms_w, float* __restrict__ yn) {
    __shared__ float sm[8];
    int row = blockIdx.x;
    const float* yr = yraw + (size_t)row * D_INNER;
    const float* zr = zx + (size_t)row * D_IN_PROJ;      // z = channels [0, D_INNER)
    float vals[6];
    float ssq = 0.f;
    #pragma unroll
    for (int i = 0; i < 6; ++i) {
        int c = threadIdx.x + i * 256;
        float v = yr[c] * silu_f(zr[c]);
        vals[i] = v;
        ssq += v * v;
    }
    float tot = block_reduce_sum(ssq, sm);
    float scale = rsqrtf(tot * (1.f / D_INNER) + 1e-5f);
    #pragma unroll
    for (int i = 0; i < 6; ++i) {
        int c = threadIdx.x + i * 256;
        yn[(size_t)row * D_INNER + c] = vals[i] * scale * rms_w[c];
    }
}

extern "C" void kernel_launch(void* const* d_in, const int* in_sizes, int n_in,
                              void* d_out, int out_size, void* d_ws, size_t ws_size,
                              hipStream_t stream) {
    const float* hidden     = (const float*)d_in[0];
    const float* ln_w       = (const float*)d_in[1];
    const float* ln_b       = (const float*)d_in[2];
    const float* in_proj_w  = (const float*)d_in[3];
    const float* conv_w     = (const float*)d_in[4];
    const float* conv_b     = (const float*)d_in[5];
    const float* dt_bias    = (const float*)d_in[6];
    const float* A_log      = (const float*)d_in[7];
    const float* Dp         = (const float*)d_in[8];
    const float* rms_w      = (const float*)d_in[9];
    const float* out_proj_w = (const float*)d_in[10];
    float* out = (float*)d_out;

    float* ws     = (float*)d_ws;
    float* normed = ws;                                   // ROWS * D_MODEL
    float* zx     = normed + (size_t)ROWS * D_MODEL;      // ROWS * D_IN_PROJ
    float* xbc    = zx + (size_t)ROWS * D_IN_PROJ;        // ROWS * CONV_DIM
    float* dtw    = xbc + (size_t)ROWS * CONV_DIM;        // ROWS * NHEADS
    float* yraw   = dtw + (size_t)ROWS * NHEADS;          // ROWS * D_INNER
    float* yn     = yraw + (size_t)ROWS * D_INNER;        // ROWS * D_INNER

    // 1) LayerNorm
    ln_kernel<<<ROWS, 256, 0, stream>>>(hidden, ln_w, ln_b, normed);

    // 2) in_proj GEMM (fp32 WMMA): zx = normed * in_proj_w^T
    wmma_gemm_f32<<<dim3((D_IN_PROJ + 63) / 64, ROWS / 128), 256, 0, stream>>>(
        normed, in_proj_w, zx, nullptr, D_MODEL, D_IN_PROJ, D_IN_PROJ);

    // 3) conv + SiLU + dt softplus
    {
        long total = (long)ROWS * (CONV_DIM + NHEADS);
        int blocks = (int)((total + 255) / 256);
        conv_dt_kernel<<<blocks, 256, 0, stream>>>(zx, conv_w, conv_b, dt_bias, xbc, dtw);
    }

    // 4) selective scan
    scan_kernel<<<BATCH * NHEADS, NSPLIT * HEADDIM, 0, stream>>>(xbc, dtw, A_log, Dp, yraw);

    // 5) gate + RMSNorm
    gate_rms_kernel<<<ROWS, 256, 0, stream>>>(yraw, zx, rms_w, yn);

    // 6) out_proj GEMM (fp32 WMMA) + residual: out = hidden + yn * out_proj_w^T
    wmma_gemm_f32<<<dim3(D_MODEL / 64, ROWS / 128), 256, 0, stream>>>(
        yn, out_proj_w, out, hidden, D_INNER, D_MODEL, D_MODEL);
}